// SGRU_89696097009944
// MI455X (gfx1250) — compile-verified
//
#include <hip/hip_runtime.h>

typedef float v2f __attribute__((ext_vector_type(2)));
typedef float v8f __attribute__((ext_vector_type(8)));

#define DEV_INLINE __device__ __forceinline__

namespace {
constexpr int T = 32, B = 8, IN = 256, H = 512, M = 1, OUT = 256;
constexpr int HM = H + M;              // 513
constexpr int NSPLIT_WH = 4;           // split-K factor for the per-step Wh GEMM
constexpr float CLIPV = 50.0f;

// ---- d_out layout (float offsets), reference return order: v,h,dU,te,tE,outs,dec
constexpr size_t OFF_V    = 0;
constexpr size_t OFF_H    = OFF_V   + (size_t)B * H;
constexpr size_t OFF_DU   = OFF_H   + (size_t)B * H;
constexpr size_t OFF_TE   = OFF_DU  + (size_t)B * H * H;
constexpr size_t OFF_TEB  = OFF_TE  + (size_t)B * H;
constexpr size_t OFF_OUTS = OFF_TEB + (size_t)B * H * H;
constexpr size_t OFF_DEC  = OFF_OUTS + (size_t)T * B * H;

// ---- workspace layout (float offsets), ~3.8 MB total
constexpr size_t W_RA  = 0;
constexpr size_t W_HI  = W_RA + (size_t)H * H;
constexpr size_t W_LO  = W_HI + (size_t)H * H;
constexpr size_t W_SV  = W_LO + (size_t)H * H;
constexpr size_t W_SE  = W_SV + H;
constexpr size_t W_SC  = W_SE + H;          // [0]=sigmoid(tau_U), [1]=sigmoid(tau_E)
constexpr size_t W_WX  = W_SC + 8;          // [T*B, HM]
constexpr size_t W_WH  = W_WX + (size_t)T * B * HM;            // [NSPLIT_WH][B, HM]
constexpr size_t W_H0  = W_WH + (size_t)NSPLIT_WH * B * HM;
constexpr size_t W_H1  = W_H0 + (size_t)B * H;
constexpr size_t W_T0  = W_H1 + (size_t)B * H;
constexpr size_t W_T1  = W_T0 + (size_t)B * H;
constexpr size_t W_VV  = W_T1 + (size_t)B * H;
constexpr size_t W_P0  = W_VV + (size_t)B * H;
constexpr size_t W_P1  = W_P0 + (size_t)B * H;
constexpr size_t W_MOD = W_P1 + (size_t)B * H;
} // namespace

DEV_INLINE float sigmf(float x) { return 1.0f / (1.0f + __expf(-x)); }
DEV_INLINE float swishf(float x) { return x * sigmf(x); }

// ---------------------------------------------------------------------------
// One-time precompute: ra = relu(alpha), clip bounds hi/lo, per-H sigmoids.
// ---------------------------------------------------------------------------
__global__ void prep_kernel(const float* __restrict__ h2h_w,
                            const float* __restrict__ alpha,
                            const float* __restrict__ tau_v,
                            const float* __restrict__ tau_e,
                            const float* __restrict__ tau_U,
                            const float* __restrict__ tau_E,
                            float* __restrict__ ra, float* __restrict__ hi,
                            float* __restrict__ lo, float* __restrict__ sv,
                            float* __restrict__ se, float* __restrict__ scal) {
  int idx = blockIdx.x * blockDim.x + threadIdx.x;
  if (idx < H * H) {
    float a = fmaxf(alpha[idx], 0.0f);
    float rae = a + 1e-8f;
    float w = h2h_w[idx];                 // rows [0,H) of h2h_w[H+M, H]
    ra[idx] = a;
    hi[idx] =  fmaxf(CLIPV - w, 0.0f) / rae;
    lo[idx] = -fmaxf(CLIPV + w, 0.0f) / rae;
  }
  if (idx < H) {
    sv[idx] = sigmf(tau_v[idx]);
    se[idx] = sigmf(tau_e[idx]);
  }
  if (idx == 0) {
    scal[0] = sigmf(tau_U[0]);
    scal[1] = sigmf(tau_E[0]);
  }
}

// ---------------------------------------------------------------------------
// Initialize small carry buffers from the inputs.
// ---------------------------------------------------------------------------
__global__ void init_small_kernel(const float* __restrict__ h0,
                                  const float* __restrict__ v0,
                                  const float* __restrict__ te0,
                                  float* __restrict__ hbuf,
                                  float* __restrict__ vbuf,
                                  float* __restrict__ tbuf) {
  int idx = blockIdx.x * blockDim.x + threadIdx.x;
  if (idx < B * H) {
    hbuf[idx] = h0[idx];
    vbuf[idx] = v0[idx];
    tbuf[idx] = te0[idx];
  }
}

// ---------------------------------------------------------------------------
// p[b,i] = sum_j ra[i,j] * dU[b,i,j] * h[b,j]   (one wave per (b,i) row)
// Used once for step 0 (dU0/h0); later steps fuse this into the plastic update.
// float4 loads: all row bases are 512-float aligned.
// ---------------------------------------------------------------------------
__global__ void matvec_p_kernel(const float* __restrict__ dU,
                                const float* __restrict__ hvec,
                                const float* __restrict__ ra,
                                float* __restrict__ p) {
  int wave = (blockIdx.x * blockDim.x + threadIdx.x) >> 5;
  int lane = threadIdx.x & 31;
  int b = wave >> 9;               // / H
  int i = wave & (H - 1);          // % H
  const float4* dU4 = (const float4*)(dU + ((size_t)b * H + i) * H);
  const float4* ra4 = (const float4*)(ra + (size_t)i * H);
  const float4* hv4 = (const float4*)(hvec + (size_t)b * H);
  float acc = 0.0f;
#pragma unroll
  for (int c = 0; c < H / 128; ++c) {
    int j4 = c * 32 + lane;
    float4 d = dU4[j4];
    float4 r = ra4[j4];
    float4 h = hv4[j4];
    acc += r.x * d.x * h.x + r.y * d.y * h.y + r.z * d.z * h.z + r.w * d.w * h.w;
  }
  for (int m = 16; m; m >>= 1) acc += __shfl_xor(acc, m, 32);
  if (lane == 0) p[b * H + i] = acc;
}

// ---------------------------------------------------------------------------
// Full-fp32 WMMA GEMM with optional split-K:
//   C[s][m,n] = sum_{k in split s} A[m,k] * W[n,k]  (+ bias[n] on split 0)
// One wave per (split, 16x16 tile) using V_WMMA_F32_16X16X4_F32.
// No load predication: out-of-range rows/cols clamp to a valid row; the
// resulting garbage lands only in C rows/cols that are never stored, so EXEC
// stays all-ones (WMMA requirement) and loads are plain global_load_b64.
//   A fragment (16x4 f32, ISA layout): lanes 0-15 -> K={k0,k0+1},
//                                      lanes 16-31 -> K={k0+2,k0+3}
//   C/D (16x16 f32): VGPR r -> M = r (+8 for upper lane half), N = lane&15
// ---------------------------------------------------------------------------
__global__ void wmma_gemm_kernel(const float* __restrict__ A,
                                 const float* __restrict__ W,
                                 const float* __restrict__ bias,
                                 float* __restrict__ C,
                                 int Ma, int N, int K, int ntn, int nsplit) {
  int wave = (blockIdx.x * blockDim.x + threadIdx.x) >> 5;
  int lane = threadIdx.x & 31;
  int mtiles = (Ma + 15) >> 4;
  int tiles = mtiles * ntn;
  if (wave >= tiles * nsplit) return;
  int s = wave / tiles;
  int tile = wave - s * tiles;
  int mt = tile / ntn;
  int nt = tile - mt * ntn;

  int half = lane >> 4;
  int l = lane & 15;
  int m = mt * 16 + l; if (m >= Ma) m = Ma - 1;   // clamp: no predication
  int n = nt * 16 + l; if (n >= N)  n = N - 1;

  int Kc = K / nsplit;
  const float* pa = A + (size_t)m * K + s * Kc + 2 * half;
  const float* pw = W + (size_t)n * K + s * Kc + 2 * half;

  v8f c = {};
#pragma unroll 4
  for (int k0 = 0; k0 < Kc; k0 += 4) {
    v2f a = *(const v2f*)(pa + k0);
    v2f b = *(const v2f*)(pw + k0);
    c = __builtin_amdgcn_wmma_f32_16x16x4_f32(false, a, false, b,
                                              (short)0, c, false, false);
  }

  int col = nt * 16 + l;
  if (col < N) {
    float bb = (bias && s == 0) ? bias[col] : 0.0f;
    float* Cs = C + (size_t)s * Ma * N;
#pragma unroll
    for (int r = 0; r < 8; ++r) {
      int row = mt * 16 + r + half * 8;
      if (row < Ma) Cs[(size_t)row * N + col] = c[r] + bb;
    }
  }
}

// ---------------------------------------------------------------------------
// Per-step small recurrence: v update, nh = swish(v), nte, mod, outs[t].
// Wh arrives as NSPLIT_WH split-K partials that are summed here.
// ---------------------------------------------------------------------------
__global__ void recurrence_kernel(const float* __restrict__ Wx_t,  // [B,HM]
                                  const float* __restrict__ Wh,    // [S][B,HM]
                                  const float* __restrict__ p,     // [B,H]
                                  const float* __restrict__ sv,
                                  const float* __restrict__ se,
                                  float* __restrict__ v,           // [B,H] in/out
                                  const float* __restrict__ h_cur,
                                  const float* __restrict__ te_cur,
                                  float* __restrict__ h_nxt,
                                  float* __restrict__ te_nxt,
                                  float* __restrict__ outs_t,      // [B,H]
                                  float* __restrict__ mod) {       // [B]
  int idx = blockIdx.x * blockDim.x + threadIdx.x;
  if (idx >= B * H) return;
  int b = idx / H;
  int i = idx - b * H;
  float wh = 0.0f;
#pragma unroll
  for (int s = 0; s < NSPLIT_WH; ++s) wh += Wh[(size_t)s * B * HM + b * HM + i];
  float dv = Wx_t[b * HM + i] + wh + p[idx];
  float svi = sv[i];
  float vv = (1.0f - svi) * v[idx] + svi * dv;
  v[idx] = vv;
  float nh = swishf(vv);
  h_nxt[idx] = nh;
  outs_t[idx] = nh;
  float sei = se[i];
  te_nxt[idx] = (1.0f - sei) * te_cur[idx] + sei * h_cur[idx];
  if (i == 0) {  // M == 1: neuromodulator is the single extra column
    float z = Wx_t[b * HM + H];
#pragma unroll
    for (int s = 0; s < NSPLIT_WH; ++s) z += Wh[(size_t)s * B * HM + b * HM + H];
    mod[b] = swishf(z);
  }
}

// ---------------------------------------------------------------------------
// Fused plastic update (the hot loop — one read + one write of dU and tE):
//   ntE = (1-sE)*tE + sE*(nh_i*te_j - nte_i*h_j)
//   dU  = clamp((1-sU)*dU + sU*mod_b*ntE, lo, hi)
//   p_next[b,i] += ra[i,j] * dU_new * nh_j         (matvec for step t+1)
// One wave owns one (b,i) row; float4 (b128) traffic; wave-reduce, no atomics.
// ---------------------------------------------------------------------------
__global__ void plastic_update_kernel(float* __restrict__ dU,
                                      float* __restrict__ tE,
                                      const float* __restrict__ ra,
                                      const float* __restrict__ hi,
                                      const float* __restrict__ lo,
                                      const float* __restrict__ h_cur,
                                      const float* __restrict__ te_cur,
                                      const float* __restrict__ h_nxt,
                                      const float* __restrict__ te_nxt,
                                      const float* __restrict__ mod,
                                      const float* __restrict__ scal,
                                      float* __restrict__ p_nxt) {
  int wave = (blockIdx.x * blockDim.x + threadIdx.x) >> 5;
  int lane = threadIdx.x & 31;
  int b = wave >> 9;
  int i = wave & (H - 1);
  const float sU = scal[0], sE = scal[1];
  const float omU = 1.0f - sU, omE = 1.0f - sE;
  const float nh_i = h_nxt[b * H + i];
  const float nte_i = te_nxt[b * H + i];
  const float sUmod = sU * mod[b];

  float4* dU4 = (float4*)(dU + ((size_t)b * H + i) * H);
  float4* tE4 = (float4*)(tE + ((size_t)b * H + i) * H);
  const float4* ra4 = (const float4*)(ra + (size_t)i * H);
  const float4* hi4 = (const float4*)(hi + (size_t)i * H);
  const float4* lo4 = (const float4*)(lo + (size_t)i * H);
  const float4* te4 = (const float4*)(te_cur + (size_t)b * H);
  const float4* hc4 = (const float4*)(h_cur + (size_t)b * H);
  const float4* nh4 = (const float4*)(h_nxt + (size_t)b * H);

  float acc = 0.0f;
#pragma unroll
  for (int c = 0; c < H / 128; ++c) {
    int j4 = c * 32 + lane;
    float4 tEv = tE4[j4];
    float4 dUv = dU4[j4];
    float4 hiv = hi4[j4];
    float4 lov = lo4[j4];
    float4 rav = ra4[j4];
    float4 tej = te4[j4];
    float4 hj  = hc4[j4];
    float4 nhj = nh4[j4];
    float4 otE, odU;
#define PLASTIC_ELT(q)                                              \
    {                                                               \
      float ntE = omE * tEv.q + sE * (nh_i * tej.q - nte_i * hj.q); \
      float nd = omU * dUv.q + sUmod * ntE;                         \
      nd = fminf(nd, hiv.q);                                        \
      nd = fmaxf(nd, lov.q);                                        \
      otE.q = ntE;                                                  \
      odU.q = nd;                                                   \
      acc += rav.q * nd * nhj.q;                                    \
    }
    PLASTIC_ELT(x) PLASTIC_ELT(y) PLASTIC_ELT(z) PLASTIC_ELT(w)
#undef PLASTIC_ELT
    tE4[j4] = otE;
    dU4[j4] = odU;
  }
  for (int m = 16; m; m >>= 1) acc += __shfl_xor(acc, m, 32);
  if (lane == 0) p_nxt[b * H + i] = acc;
}

// ---------------------------------------------------------------------------
extern "C" void kernel_launch(void* const* d_in, const int* in_sizes, int n_in,
                              void* d_out, int out_size, void* d_ws,
                              size_t ws_size, hipStream_t stream) {
  (void)in_sizes; (void)n_in; (void)out_size; (void)ws_size;
  const float* x     = (const float*)d_in[0];
  const float* h0    = (const float*)d_in[1];
  const float* v0    = (const float*)d_in[2];
  const float* dU0   = (const float*)d_in[3];
  const float* te0   = (const float*)d_in[4];
  const float* tE0   = (const float*)d_in[5];
  const float* x2h_w = (const float*)d_in[6];
  const float* x2h_b = (const float*)d_in[7];
  const float* h2h_w = (const float*)d_in[8];
  const float* h2h_b = (const float*)d_in[9];
  const float* alpha = (const float*)d_in[10];
  const float* tau_v = (const float*)d_in[11];
  const float* tau_e = (const float*)d_in[12];
  const float* tau_U = (const float*)d_in[13];
  const float* tau_E = (const float*)d_in[14];
  const float* dec_w = (const float*)d_in[15];
  const float* dec_b = (const float*)d_in[16];

  float* out = (float*)d_out;
  float* ws = (float*)d_ws;

  float* dU = out + OFF_DU;      // updated in place in d_out
  float* tE = out + OFF_TEB;     // updated in place in d_out
  float* outs = out + OFF_OUTS;

  hipMemcpyAsync(dU, dU0, sizeof(float) * (size_t)B * H * H,
                 hipMemcpyDeviceToDevice, stream);
  hipMemcpyAsync(tE, tE0, sizeof(float) * (size_t)B * H * H,
                 hipMemcpyDeviceToDevice, stream);

  prep_kernel<<<(H * H + 255) / 256, 256, 0, stream>>>(
      h2h_w, alpha, tau_v, tau_e, tau_U, tau_E,
      ws + W_RA, ws + W_HI, ws + W_LO, ws + W_SV, ws + W_SE, ws + W_SC);
  init_small_kernel<<<(B * H + 255) / 256, 256, 0, stream>>>(
      h0, v0, te0, ws + W_H0, ws + W_VV, ws + W_T0);
  matvec_p_kernel<<<B * H / 8, 256, 0, stream>>>(dU0, h0, ws + W_RA, ws + W_P0);

  // Wx for ALL timesteps in one WMMA GEMM: [T*B,HM] = x[T*B,IN] @ x2h_w^T + b
  // 16 x 33 tiles = 528 waves -> 66 blocks of 8 waves.
  wmma_gemm_kernel<<<66, 256, 0, stream>>>(x, x2h_w, x2h_b, ws + W_WX,
                                           T * B, HM, IN, (HM + 15) / 16, 1);

  float* h_cur = ws + W_H0; float* h_nxt = ws + W_H1;
  float* t_cur = ws + W_T0; float* t_nxt = ws + W_T1;
  float* p_cur = ws + W_P0; float* p_nxt = ws + W_P1;

  // 33 tiles x 4 K-splits = 132 waves -> 17 blocks of 8 waves (tail guarded)
  constexpr int WH_WAVES = 33 * NSPLIT_WH;
  constexpr int WH_BLOCKS = (WH_WAVES * 32 + 255) / 256;

  for (int t = 0; t < T; ++t) {
    wmma_gemm_kernel<<<WH_BLOCKS, 256, 0, stream>>>(
        h_cur, h2h_w, h2h_b, ws + W_WH, B, HM, H, (HM + 15) / 16, NSPLIT_WH);
    recurrence_kernel<<<(B * H + 255) / 256, 256, 0, stream>>>(
        ws + W_WX + (size_t)t * B * HM, ws + W_WH, p_cur, ws + W_SV, ws + W_SE,
        ws + W_VV, h_cur, t_cur, h_nxt, t_nxt, outs + (size_t)t * B * H,
        ws + W_MOD);
    plastic_update_kernel<<<B * H / 8, 256, 0, stream>>>(
        dU, tE, ws + W_RA, ws + W_HI, ws + W_LO, h_cur, t_cur, h_nxt, t_nxt,
        ws + W_MOD, ws + W_SC, p_nxt);
    float* tmp;
    tmp = h_cur; h_cur = h_nxt; h_nxt = tmp;
    tmp = t_cur; t_cur = t_nxt; t_nxt = tmp;
    tmp = p_cur; p_cur = p_nxt; p_nxt = tmp;
  }

  hipMemcpyAsync(out + OFF_V, ws + W_VV, sizeof(float) * B * H,
                 hipMemcpyDeviceToDevice, stream);
  hipMemcpyAsync(out + OFF_H, h_cur, sizeof(float) * B * H,
                 hipMemcpyDeviceToDevice, stream);
  hipMemcpyAsync(out + OFF_TE, t_cur, sizeof(float) * B * H,
                 hipMemcpyDeviceToDevice, stream);

  // Decoder: dec[T*B,OUT] = outs[T*B,H] @ dec_w^T + dec_b
  // 16 x 16 tiles = 256 waves -> 32 blocks of 8 waves.
  wmma_gemm_kernel<<<32, 256, 0, stream>>>(outs, dec_w, dec_b, out + OFF_DEC,
                                           T * B, OUT, H, OUT / 16, 1);
}